// BattleModel_70171175682733
// MI455X (gfx1250) — compile-verified
//
#include <hip/hip_runtime.h>
#include <hip/hip_bf16.h>

// ---------------------------------------------------------------------------
// BattleModel for MI455X (gfx1250, wave32).
//
// Stage 1 (mlp_pool): per-unit GEMM [N,9]x[9,64] via V_WMMA_F32_16X16X4_F32
//   (K padded 9->12, 3 K-steps x 4 N-tiles per 16-unit tile), ReLU, then
//   sorted-segment sum with run-length-compressed global f32 atomics.
//   Feature rows stream into LDS via GLOBAL_LOAD_ASYNC_TO_LDS_B32 (ASYNCcnt).
// Stage 2 (head): [B,128]x[128,32] via WMMA (32 K-steps x 2 N-tiles), ReLU,
//   dot with Wc2 via cross-lane shfl reduction, sigmoid. Combined tile staged
//   with GLOBAL_LOAD_ASYNC_TO_LDS_B128.
// Weight LDS layouts are K-pair-packed so every WMMA B fragment is a single
// 8B-aligned ds_load_b64 into an even VGPR pair (no fix-up moves).
// ---------------------------------------------------------------------------

#define N_UNITS 524288
#define BATCH   16384

typedef float v2f __attribute__((ext_vector_type(2)));
typedef float v8f __attribute__((ext_vector_type(8)));

__device__ __forceinline__ v8f wmma_f32_k4(v2f a, v2f b, v8f c) {
    // 8 args: (neg_a, A, neg_b, B, c_mod, C, reuse_a, reuse_b)
    return __builtin_amdgcn_wmma_f32_16x16x4_f32(false, a, false, b,
                                                 (short)0, c, false, false);
}

// gfx1250 async global->LDS copies (tracked by ASYNCcnt, no VGPR data).
__device__ __forceinline__ void async_load_b32(unsigned lds_off, const void* gaddr) {
    asm volatile("global_load_async_to_lds_b32 %0, %1, off"
                 :: "v"(lds_off), "v"(gaddr) : "memory");
}
__device__ __forceinline__ void async_load_b128(unsigned lds_off, const void* gaddr) {
    asm volatile("global_load_async_to_lds_b128 %0, %1, off"
                 :: "v"(lds_off), "v"(gaddr) : "memory");
}
__device__ __forceinline__ void wait_asynccnt0() {
    asm volatile("s_wait_asynccnt 0x0" ::: "memory");
}
__device__ __forceinline__ unsigned lds_off(const void* p) {
    return (unsigned)(unsigned long long)p;   // AS(3) pointer == LDS byte offset
}

// ---------------------------------------------------------------------------
// Zero the pooled accumulators in d_ws (2 sides x BATCH x 64 floats).
// ---------------------------------------------------------------------------
__global__ __launch_bounds__(256) void zero_ws_kernel(float4* __restrict__ p, int n4) {
    int i = blockIdx.x * blockDim.x + threadIdx.x;
    if (i < n4) p[i] = make_float4(0.f, 0.f, 0.f, 0.f);
}

// ---------------------------------------------------------------------------
// Stage 1: per-unit MLP (9->64, ReLU) + sorted segment-sum pooling.
// Block = 256 threads = 8 waves; each wave owns one 16-unit tile.
// grid.y = side (0: left, 1: right).
// ---------------------------------------------------------------------------
__global__ __launch_bounds__(256) void mlp_pool_kernel(
    const float* __restrict__ feats_l, const float* __restrict__ feats_r,
    const int*   __restrict__ seg_l,   const int*   __restrict__ seg_r,
    const float* __restrict__ W1,      const float* __restrict__ b1,
    float* __restrict__ pooled_base)
{
    const int side = blockIdx.y;
    const float* feats = side ? feats_r : feats_l;
    const int*   seg   = side ? seg_r   : seg_l;
    float* pooled = pooled_base + side * (BATCH * 64);

    // K-pair-packed W1: ldsW[kpair*128 + col*2 + j] = W1[kpair*2+j][col] (0-padded K>=9)
    __shared__ __align__(16) float ldsW[6 * 128];
    __shared__ __align__(16) float ldsF[8][16 * 12];   // per-wave feats, padded K 9->12

    const int tid  = threadIdx.x;
    const int wave = tid >> 5;
    const int lane = tid & 31;
    const int half = lane >> 4;
    const int lm   = lane & 15;

    const int tile = blockIdx.x * 8 + wave;
    const int u0   = tile * 16;

    // Async-stream 16 feature rows (144 contiguous floats) into LDS.
#pragma unroll
    for (int base = 0; base < 144; base += 32) {
        int idx = base + lane;
        if (idx < 144) {
            int r = idx / 9, k = idx - r * 9;
            async_load_b32(lds_off(&ldsF[wave][r * 12 + k]), feats + u0 * 9 + idx);
        }
    }
    // Zero the K-padding slots (k = 9..11 of each row).
    if (lane < 16) {
        int b = lane * 12;
        ldsF[wave][b + 9]  = 0.f;
        ldsF[wave][b + 10] = 0.f;
        ldsF[wave][b + 11] = 0.f;
    }

    // Stage W1 (9x64) K-pair-packed, zero-padding K rows 9..11.
    for (int idx = tid; idx < 6 * 64; idx += 256) {
        int pr = idx >> 6, col = idx & 63;
        int k0 = pr * 2;
        v2f w;
        w.x = (k0     < 9) ? W1[k0 * 64 + col]       : 0.f;
        w.y = (k0 + 1 < 9) ? W1[(k0 + 1) * 64 + col] : 0.f;
        *(v2f*)&ldsW[pr * 128 + col * 2] = w;
    }

    // Segment ids: lane j holds seg[u0 + (j%16) + 8*(j/16)] so that
    // shfl(width=16, src=r) delivers seg of row r+8*half to every lane.
    int srow = u0 + lm + 8 * half;
    if (srow > N_UNITS - 1) srow = N_UNITS - 1;
    int segv = seg[srow];

    wait_asynccnt0();
    __syncthreads();

    int segm[8];
#pragma unroll
    for (int r = 0; r < 8; r++) segm[r] = __shfl(segv, r, 16);

    // GEMM: 3 K-steps (K=12 padded) x 4 N-tiles of 16 columns.
    // A/B fragments: VGPR pair holds K {kk, kk+1}, halves carry K-offset 0 / 2.
    v8f acc[4] = {};
#pragma unroll
    for (int ks = 0; ks < 3; ks++) {
        const int kk = ks * 4 + 2 * half;
        const v2f a = *(const v2f*)&ldsF[wave][lm * 12 + kk];
        const int pr = ks * 2 + half;
#pragma unroll
        for (int nt = 0; nt < 4; nt++) {
            const int col = nt * 16 + lm;
            const v2f b = *(const v2f*)&ldsW[pr * 128 + col * 2];
            acc[nt] = wmma_f32_k4(a, b, acc[nt]);
        }
    }

    // Bias + ReLU, then run-length-compressed segment-sum atomics.
    // C layout: vgpr r, lane -> row (r + 8*half), col (nt*16 + lm).
#pragma unroll
    for (int nt = 0; nt < 4; nt++) {
        const int col = nt * 16 + lm;
        const float bias = b1[col];
        float h[8];
#pragma unroll
        for (int r = 0; r < 8; r++) {
            float v = acc[nt][r] + bias;
            h[r] = v > 0.f ? v : 0.f;
        }
        int   runSeg = segm[0];
        float runSum = h[0];
#pragma unroll
        for (int r = 1; r < 8; r++) {
            if (segm[r] == runSeg) {
                runSum += h[r];
            } else {
                atomicAdd(&pooled[runSeg * 64 + col], runSum);
                runSeg = segm[r];
                runSum = h[r];
            }
        }
        atomicAdd(&pooled[runSeg * 64 + col], runSum);
    }
}

// ---------------------------------------------------------------------------
// Stage 2: head. combined[B,128] @ Wc1[128,32] + bc1, ReLU, @ Wc2[32] + bc2,
// sigmoid. Block = 128 threads = 4 waves; each wave owns a 16-row tile.
// ---------------------------------------------------------------------------
__global__ __launch_bounds__(128) void head_kernel(
    const float* __restrict__ pooled_base,
    const float* __restrict__ Wc1, const float* __restrict__ bc1,
    const float* __restrict__ Wc2, const float* __restrict__ bc2,
    float* __restrict__ out)
{
    // K-pair-packed Wc1: ldsW[kpair*64 + col*2 + j] = Wc1[kpair*2+j][col]
    __shared__ __align__(16) float ldsW[64 * 64];      // 16 KB
    __shared__ __align__(16) float ldsC[4][16 * 128];  // 32 KB, per-wave combined tile

    const int tid  = threadIdx.x;
    const int wave = tid >> 5;
    const int lane = tid & 31;
    const int half = lane >> 4;
    const int lm   = lane & 15;

    const float* L = pooled_base;
    const float* R = pooled_base + BATCH * 64;

    const int tile = blockIdx.x * 4 + wave;
    const int u0   = tile * 16;

    // Async-stage combined tile: 16 rows x 128 f32 = 512 float4, coalesced.
#pragma unroll
    for (int i = 0; i < 16; i++) {
        int idx = i * 32 + lane;
        int row = idx >> 5;          // 0..15
        int c4  = idx & 31;          // float4 index within the 128-wide row
        const float* src = (c4 < 16) ? (L + (u0 + row) * 64 + c4 * 4)
                                     : (R + (u0 + row) * 64 + (c4 - 16) * 4);
        async_load_b128(lds_off(&ldsC[wave][row * 128 + c4 * 4]), src);
    }

    // Stage Wc1 K-pair-packed.
    for (int idx = tid; idx < 64 * 32; idx += 128) {
        int pr = idx >> 5, col = idx & 31;
        int k0 = pr * 2;
        v2f w;
        w.x = Wc1[k0 * 32 + col];
        w.y = Wc1[(k0 + 1) * 32 + col];
        *(v2f*)&ldsW[pr * 64 + col * 2] = w;
    }

    wait_asynccnt0();
    __syncthreads();

    // GEMM: 32 K-steps x 2 N-tiles (N=32).
    v8f acc0 = {}, acc1 = {};
#pragma unroll 8
    for (int ks = 0; ks < 32; ks++) {
        const int kk = ks * 4 + 2 * half;
        const int pr = ks * 2 + half;
        const v2f a   = *(const v2f*)&ldsC[wave][lm * 128 + kk];
        const v2f b0  = *(const v2f*)&ldsW[pr * 64 + lm * 2];
        const v2f b1v = *(const v2f*)&ldsW[pr * 64 + (16 + lm) * 2];
        acc0 = wmma_f32_k4(a, b0, acc0);
        acc1 = wmma_f32_k4(a, b1v, acc1);
    }

    // ReLU + dot with Wc2 + bc2 + sigmoid.
    const float bias0 = bc1[lm],      bias1 = bc1[16 + lm];
    const float w0    = Wc2[lm],      w1    = Wc2[16 + lm];
    const float bb    = bc2[0];

#pragma unroll
    for (int r = 0; r < 8; r++) {
        float h0 = acc0[r] + bias0; h0 = h0 > 0.f ? h0 : 0.f;
        float h1 = acc1[r] + bias1; h1 = h1 > 0.f ? h1 : 0.f;
        float pr = h0 * w0 + h1 * w1;
        // Reduce across the 16 lanes of this half (row = r + 8*half).
        pr += __shfl_xor(pr, 1, 16);
        pr += __shfl_xor(pr, 2, 16);
        pr += __shfl_xor(pr, 4, 16);
        pr += __shfl_xor(pr, 8, 16);
        if (lm == r) {
            float logit = pr + bb;
            out[u0 + r + 8 * half] = 1.f / (1.f + __expf(-logit));
        }
    }
}

// ---------------------------------------------------------------------------
extern "C" void kernel_launch(void* const* d_in, const int* in_sizes, int n_in,
                              void* d_out, int out_size, void* d_ws, size_t ws_size,
                              hipStream_t stream) {
    const float* feats_l = (const float*)d_in[0];
    const float* feats_r = (const float*)d_in[1];
    const int*   seg_l   = (const int*)d_in[2];
    const int*   seg_r   = (const int*)d_in[3];
    const float* W1      = (const float*)d_in[4];
    const float* b1      = (const float*)d_in[5];
    const float* Wc1     = (const float*)d_in[6];
    const float* bc1     = (const float*)d_in[7];
    const float* Wc2     = (const float*)d_in[8];
    const float* bc2     = (const float*)d_in[9];
    float* out    = (float*)d_out;
    float* pooled = (float*)d_ws;   // 2 * BATCH * 64 floats = 8 MB

    const int n4 = (2 * BATCH * 64) / 4;
    zero_ws_kernel<<<(n4 + 255) / 256, 256, 0, stream>>>((float4*)pooled, n4);

    dim3 grid1(N_UNITS / 16 / 8, 2);   // 4096 x 2 blocks, 8 tiles/block
    mlp_pool_kernel<<<grid1, 256, 0, stream>>>(feats_l, feats_r, seg_l, seg_r,
                                               W1, b1, pooled);

    head_kernel<<<BATCH / 16 / 4, 128, 0, stream>>>(pooled, Wc1, bc1, Wc2, bc2, out);
}